// TTTLayer_53970559041675
// MI455X (gfx1250) — compile-verified
//
#include <hip/hip_runtime.h>

typedef float v2f __attribute__((ext_vector_type(2)));
typedef float v8f __attribute__((ext_vector_type(8)));

#define WMMA_F32_16x16x4(a, b, c) \
    __builtin_amdgcn_wmma_f32_16x16x4_f32(false, (a), false, (b), (short)0, (c), false, false)

// Async global -> LDS copy of 16 bytes (CDNA5 ASYNCcnt path, no VGPR staging).
__device__ __forceinline__ void async_copy_b128(const void* gptr, void* lptr) {
    unsigned long long g = (unsigned long long)(uintptr_t)gptr;
    unsigned l = (unsigned)(uintptr_t)lptr;   // low 32 bits of flat LDS addr == LDS offset
    asm volatile("global_load_async_to_lds_b128 %0, %1, off" :: "v"(l), "v"(g) : "memory");
}
__device__ __forceinline__ void wait_async0() {
    asm volatile("s_wait_asynccnt 0x0" ::: "memory");
}

// ---------------------------------------------------------------------------
// TT contraction, stored TRANSPOSED (n-major) for LDS-friendly B fragments:
// G01t[r2][p=n0*64+n1] = sum_r1 core0[n0,r1]*core1[r1,n1,r2]   -> (16, 4096)
// ---------------------------------------------------------------------------
__global__ void tt_build_g01t(const float* __restrict__ core0,
                              const float* __restrict__ core1,
                              float* __restrict__ G01t) {
    int g  = blockIdx.x * 256 + threadIdx.x;   // 65536 outputs
    int r2 = g >> 12;
    int p  = g & 4095;
    int n0 = p >> 6;
    int n1 = p & 63;
    float s = 0.f;
#pragma unroll
    for (int r1 = 0; r1 < 16; ++r1)
        s += core0[n0 * 16 + r1] * core1[(r1 * 64 + n1) * 16 + r2];
    G01t[g] = s;                               // == G01t[r2*4096 + p]
}

// G23[r2][q=n2*64+n3] = sum_r3 core2[r2,n2,r3]*core3[r3,n3,0]   -> (16, 4096)
__global__ void tt_build_g23(const float* __restrict__ core2,
                             const float* __restrict__ core3,
                             float* __restrict__ G23) {
    int g  = blockIdx.x * 256 + threadIdx.x;   // 65536 outputs
    int r2 = g >> 12;
    int q  = g & 4095;
    int n2 = q >> 6;
    int n3 = q & 63;
    float s = 0.f;
#pragma unroll
    for (int r3 = 0; r3 < 16; ++r3)
        s += core2[(r2 * 64 + n2) * 16 + r3] * core3[r3 * 64 + n3];
    G23[g] = s;                                // == G23[r2*4096 + q]
}

// ---------------------------------------------------------------------------
// T = x @ G01   (8192x4096 @ 4096x16), K split 8 ways -> T8[8][8192][16]
// Block: 256 threads (8 waves). BM=128 (16 rows per wave), K-slice 512,
// staged 64 K at a time via async global->LDS copies, double-buffered so the
// next stage's HBM reads overlap the current stage's WMMAs.
// Row stride 68 in LDS -> conflict-free per-lane float2 reads for BOTH the
// A fragments (x rows) and B fragments (G01t rows).
// ---------------------------------------------------------------------------
__global__ void tt_xg(const float* __restrict__ x,
                      const float* __restrict__ G01t,
                      float* __restrict__ T8) {
    __shared__ __align__(16) float Xs[2][128 * 68];
    __shared__ __align__(16) float Gs[2][16 * 68];

    const int tid  = threadIdx.x;
    const int lane = tid & 31;
    const int wave = tid >> 5;
    const int hw   = lane >> 4;        // half-wave: 0 or 1
    const int lm   = lane & 15;
    const int m0   = blockIdx.x * 128;
    const int kb   = blockIdx.y * 512;

    auto issue_stage = [&](int st, int buf) {
        const int k0 = kb + st * 64;
        // x tile: 128 rows x 64 cols, 8 async b128 per thread
#pragma unroll
        for (int i = 0; i < 8; ++i) {
            int idx = tid + i * 256;
            int row = idx >> 4;
            int c4  = idx & 15;
            async_copy_b128(x + (size_t)(m0 + row) * 4096 + k0 + c4 * 4,
                            &Xs[buf][row * 68 + c4 * 4]);
        }
        // G01t slice: 16 rows x 64 cols, 1 async b128 per thread
        {
            int row = tid >> 4;        // 0..15  (n index)
            int c4  = tid & 15;        // 16 float4 per row
            async_copy_b128(G01t + (size_t)row * 4096 + k0 + c4 * 4,
                            &Gs[buf][row * 68 + c4 * 4]);
        }
    };

    v8f acc;
#pragma unroll
    for (int i = 0; i < 8; ++i) acc[i] = 0.f;

    issue_stage(0, 0);
    for (int st = 0; st < 8; ++st) {
        const int buf = st & 1;
        wait_async0();         // my copies for stage st are in LDS
        __syncthreads();       // everyone's are; all waves done reading buf^1
        if (st < 7) issue_stage(st + 1, buf ^ 1);

        const float* Xp = &Xs[buf][0];
        const float* Gp = &Gs[buf][0];
        const int arow = (wave * 16 + lm) * 68 + hw * 2;
        const int brow = lm * 68 + hw * 2;
#pragma unroll
        for (int kk = 0; kk < 64; kk += 4) {
            v2f a = *(const v2f*)(Xp + arow + kk);   // A[m][k..k+1]
            v2f b = *(const v2f*)(Gp + brow + kk);   // B[k..k+1][n] = G01t[n][k..k+1]
            acc = WMMA_F32_16x16x4(a, b, acc);
        }
    }

    float* Tp = T8 + (size_t)blockIdx.y * 131072;    // 8192*16 per slice
#pragma unroll
    for (int r = 0; r < 8; ++r)
        Tp[(m0 + wave * 16 + hw * 8 + r) * 16 + lm] = acc[r];
}

// Sum the 8 K-split partials: T[idx] = sum_s T8[s][idx]
__global__ void tt_reduce_t(const float* __restrict__ T8, float* __restrict__ T) {
    int idx = blockIdx.x * 256 + threadIdx.x;        // 131072 outputs
    float s = 0.f;
#pragma unroll
    for (int p = 0; p < 8; ++p) s += T8[(size_t)p * 131072 + idx];
    T[idx] = s;
}

// ---------------------------------------------------------------------------
// y = T @ G23   (8192x16 @ 16x4096). Block: 8 waves; each wave computes a
// 64x16 output strip (4 M-subtiles, K=16 = 4 WMMA steps, B frags reused 4x).
// T (512KB) and G23 (256KB) stay L2-resident. The 64x128 block tile is staged
// through LDS (row stride 132, conflict-free) so y is written as block-wide
// float4 rows: 32 consecutive lanes -> 512B fully contiguous stores.
// ---------------------------------------------------------------------------
__global__ void tt_yg(const float* __restrict__ T,
                      const float* __restrict__ G23,
                      float* __restrict__ y) {
    __shared__ __align__(16) float Cs[64 * 132];

    const int tid  = threadIdx.x;
    const int lane = tid & 31;
    const int wave = tid >> 5;
    const int hw   = lane >> 4;
    const int lm   = lane & 15;
    const int m0   = blockIdx.x * 64;
    const int n0   = blockIdx.y * 128;
    const int nn   = n0 + wave * 16 + lm;

    v2f b[4];
#pragma unroll
    for (int t = 0; t < 4; ++t) {
        int k = t * 4 + hw * 2;
        b[t][0] = G23[(size_t)(k + 0) * 4096 + nn];
        b[t][1] = G23[(size_t)(k + 1) * 4096 + nn];
    }

    v8f acc[4];
#pragma unroll
    for (int mt = 0; mt < 4; ++mt)
#pragma unroll
        for (int i = 0; i < 8; ++i) acc[mt][i] = 0.f;

#pragma unroll
    for (int mt = 0; mt < 4; ++mt) {
        const float* Trow = T + (size_t)(m0 + mt * 16 + lm) * 16 + hw * 2;
#pragma unroll
        for (int t = 0; t < 4; ++t) {
            v2f a = *(const v2f*)(Trow + t * 4);
            acc[mt] = WMMA_F32_16x16x4(a, b[t], acc[mt]);
        }
    }

    // stage the block's 64x128 tile in LDS, then write coalesced rows
#pragma unroll
    for (int mt = 0; mt < 4; ++mt)
#pragma unroll
        for (int r = 0; r < 8; ++r)
            Cs[(mt * 16 + hw * 8 + r) * 132 + wave * 16 + lm] = acc[mt][r];
    __syncthreads();

#pragma unroll
    for (int i = 0; i < 8; ++i) {
        int idx = tid + i * 256;           // 2048 float4 = 64 rows x 128 cols
        int row = idx >> 5;
        int c4  = idx & 31;
        float4 v = *(const float4*)(&Cs[row * 132 + c4 * 4]);
        *(float4*)(y + (size_t)(m0 + row) * 4096 + n0 + c4 * 4) = v;
    }
}

// ---------------------------------------------------------------------------
extern "C" void kernel_launch(void* const* d_in, const int* in_sizes, int n_in,
                              void* d_out, int out_size, void* d_ws, size_t ws_size,
                              hipStream_t stream) {
    (void)in_sizes; (void)n_in; (void)out_size; (void)ws_size;
    const float* x  = (const float*)d_in[0];
    const float* c0 = (const float*)d_in[1];
    const float* c1 = (const float*)d_in[2];
    const float* c2 = (const float*)d_in[3];
    const float* c3 = (const float*)d_in[4];
    float* y = (float*)d_out;

    char* ws    = (char*)d_ws;
    float* G01t = (float*)(ws);                       // 16x4096  f32 = 256 KB
    float* G23  = (float*)(ws + (262144));            // 16x4096  f32 = 256 KB
    float* T8   = (float*)(ws + (524288));            // 8x8192x16 f32 = 4 MB
    float* T    = (float*)(ws + (524288 + 4194304));  // 8192x16  f32 = 512 KB

    tt_build_g01t<<<256, 256, 0, stream>>>(c0, c1, G01t);
    tt_build_g23<<<256, 256, 0, stream>>>(c2, c3, G23);
    tt_xg<<<dim3(64, 8), 256, 0, stream>>>(x, G01t, T8);
    tt_reduce_t<<<512, 256, 0, stream>>>(T8, T);
    tt_yg<<<dim3(128, 32), 256, 0, stream>>>(T, G23, y);
}